// HyperbolicGraphConv_2508260901395
// MI455X (gfx1250) — compile-verified
//
#include <hip/hip_runtime.h>
#include <hip/hip_bf16.h>

typedef __attribute__((ext_vector_type(2))) float v2f;
typedef __attribute__((ext_vector_type(8))) float v8f;

#define DIM 64
#define HEPS 1e-7f
#define HMAX_NORM (1.0f - 1e-5f)

// ---------------------------------------------------------------------------
// prep: w_h = expmap0(weight) row-wise, b_h = expmap0(bias), y2 = ||b_h||^2
// ---------------------------------------------------------------------------
__global__ __launch_bounds__(64) void hgc_prep_kernel(
    const float* __restrict__ weight, const float* __restrict__ bias,
    float* __restrict__ wh, float* __restrict__ bh, float* __restrict__ y2) {
  int i = threadIdx.x;  // 0..63, one row of weight each
  float ss = 0.0f;
  for (int j = 0; j < DIM; ++j) { float w = weight[i * DIM + j]; ss += w * w; }
  float n = fmaxf(sqrtf(ss), HEPS);
  float sc = tanhf(n) / n;
  for (int j = 0; j < DIM; ++j) wh[i * DIM + j] = weight[i * DIM + j] * sc;
  if (i == 0) {
    float sb = 0.0f;
    for (int j = 0; j < DIM; ++j) { float b = bias[j]; sb += b * b; }
    float nb = fmaxf(sqrtf(sb), HEPS);
    float sbc = tanhf(nb) / nb;
    float yy = 0.0f;
    for (int j = 0; j < DIM; ++j) { float v = bias[j] * sbc; bh[j] = v; yy += v * v; }
    *y2 = yy;
  }
}

// ---------------------------------------------------------------------------
// transform: per 16-node tile, Mx = x @ w_h via v_wmma_f32_16x16x4_f32,
// then mobius_matvec scaling + mobius_add(b_h) epilogue. 128 threads = 4 waves.
// ---------------------------------------------------------------------------
__global__ __launch_bounds__(128) void hgc_transform_kernel(
    const float* __restrict__ x, const float* __restrict__ wh,
    const float* __restrict__ bh, const float* __restrict__ y2p,
    float* __restrict__ h, int N) {
  __shared__ float swh[DIM][DIM + 1];   // w_h, padded
  __shared__ float sx[16][DIM + 1];     // x tile
  __shared__ float smx[16][DIM + 1];    // Mx tile
  __shared__ float sbh[DIM];

  const int tid = threadIdx.x;
  const int r0 = blockIdx.x * 16;

  for (int i = tid; i < DIM * DIM; i += 128) swh[i >> 6][i & 63] = wh[i];
  for (int i = tid; i < 16 * DIM; i += 128) {
    int row = r0 + (i >> 6);
    sx[i >> 6][i & 63] = (row < N) ? x[(size_t)row * DIM + (i & 63)] : 0.0f;
  }
  if (tid < DIM) sbh[tid] = bh[tid];
  __syncthreads();

  const int lane = tid & 31;
  const int wave = tid >> 5;     // 4 waves -> 4 column tiles of 16
  const int hi   = lane >> 4;    // lane half (K offset select)
  const int l15  = lane & 15;
  const int c0   = wave * 16;

  // C/D f32 16x16: VGPR i -> M = i + 8*hi, N = l15
  v8f c = {0.f, 0.f, 0.f, 0.f, 0.f, 0.f, 0.f, 0.f};
  for (int k = 0; k < DIM; k += 4) {
    v2f a, b;
    // A 16x4 f32: v0 = K (k + 2*hi), v1 = K (k + 2*hi + 1), row = l15
    a.x = sx[l15][k + 2 * hi];
    a.y = sx[l15][k + 2 * hi + 1];
    // B 4x16 f32: mirrored layout, col = c0 + l15
    b.x = swh[k + 2 * hi][c0 + l15];
    b.y = swh[k + 2 * hi + 1][c0 + l15];
    c = __builtin_amdgcn_wmma_f32_16x16x4_f32(
        /*neg_a=*/false, a, /*neg_b=*/false, b,
        /*c_mod=*/(short)0, c, /*reuse_a=*/false, /*reuse_b=*/false);
  }
#pragma unroll
  for (int i = 0; i < 8; ++i) smx[i + 8 * hi][c0 + l15] = c[i];
  __syncthreads();

  // per-row epilogue: 8 threads per row, each owns 8 columns
  const int row = tid >> 3;
  const int g = tid & 7;
  float sxx = 0.f, smm = 0.f, smb = 0.f;
  float mv[8], bv[8];
#pragma unroll
  for (int j = 0; j < 8; ++j) {
    int col = g * 8 + j;
    float xv = sx[row][col];
    mv[j] = smx[row][col];
    bv[j] = sbh[col];
    sxx += xv * xv;
    smm += mv[j] * mv[j];
    smb += mv[j] * bv[j];
  }
  // reduce across the 8 contiguous lanes of this row group (wave32 shuffles)
  for (int m = 1; m < 8; m <<= 1) {
    sxx += __shfl_xor(sxx, m, 32);
    smm += __shfl_xor(smm, m, 32);
    smb += __shfl_xor(smb, m, 32);
  }
  float xn  = fmaxf(sqrtf(sxx), HEPS);
  float mxn = fmaxf(sqrtf(smm), HEPS);
  float xc  = fminf(xn, 1.0f - 1e-7f);
  float art = 0.5f * log1pf(2.0f * xc / (1.0f - xc));     // artanh(xn)
  float t   = tanhf(mxn / xn * art);
  float s   = t / mxn;                                    // h1 = s * Mx
  float x2  = s * s * smm;                                // ||h1||^2
  float xy  = s * smb;                                    // h1 . b_h
  float yy  = *y2p;                                       // ||b_h||^2
  float den = fmaxf(1.0f + 2.0f * xy + x2 * yy, HEPS);
  float ch  = (1.0f + 2.0f * xy + yy) / den;
  float cb  = (1.0f - x2) / den;
  int grow = r0 + row;
  if (grow < N) {
    float* hp = h + (size_t)grow * DIM + g * 8;
#pragma unroll
    for (int j = 0; j < 8; ++j) hp[j] = ch * s * mv[j] + cb * bv[j];
  }
}

// ---------------------------------------------------------------------------
__global__ void hgc_zero_kernel(float* __restrict__ p, int n) {
  int i = blockIdx.x * blockDim.x + threadIdx.x;
  if (i < n) p[i] = 0.0f;
}

// edge scatter: one wave32 per edge, lane owns 2 features -> coalesced
// gather of h[src] and coalesced global_atomic_add_f32 into out[dst]
__global__ __launch_bounds__(256) void hgc_edge_kernel(
    const float* __restrict__ h, const int* __restrict__ src,
    const int* __restrict__ dst, float* __restrict__ agg,
    float* __restrict__ deg, int E) {
  int wid = (int)((blockIdx.x * (size_t)blockDim.x + threadIdx.x) >> 5);
  int lane = threadIdx.x & 31;
  if (wid >= E) return;
  int s = src[wid];
  int d = dst[wid];
  const float2* hp = (const float2*)(h + (size_t)s * DIM);
  float2 v = hp[lane];
  float* ap = agg + (size_t)d * DIM + lane * 2;
  atomicAdd(ap, v.x);
  atomicAdd(ap + 1, v.y);
  if (lane == 0) atomicAdd(deg + d, 1.0f);
}

// finalize: one wave32 per node -> mean + project into unit ball
__global__ __launch_bounds__(256) void hgc_final_kernel(
    float* __restrict__ out, const float* __restrict__ deg, int N) {
  int wid = (int)((blockIdx.x * (size_t)blockDim.x + threadIdx.x) >> 5);
  int lane = threadIdx.x & 31;
  if (wid >= N) return;
  float inv = 1.0f / fmaxf(deg[wid], 1.0f);
  float2* op = (float2*)(out + (size_t)wid * DIM);
  float2 v = op[lane];
  v.x *= inv; v.y *= inv;
  float ss = v.x * v.x + v.y * v.y;
  for (int m = 1; m < 32; m <<= 1) ss += __shfl_xor(ss, m, 32);
  float n = fmaxf(sqrtf(ss), HEPS);
  float sc = (n > HMAX_NORM) ? (HMAX_NORM / n) : 1.0f;
  v.x *= sc; v.y *= sc;
  op[lane] = v;
}

// ---------------------------------------------------------------------------
extern "C" void kernel_launch(void* const* d_in, const int* in_sizes, int n_in,
                              void* d_out, int out_size, void* d_ws, size_t ws_size,
                              hipStream_t stream) {
  (void)n_in; (void)out_size; (void)ws_size;
  const float* x      = (const float*)d_in[0];
  const float* weight = (const float*)d_in[1];
  const float* bias   = (const float*)d_in[2];
  const int*   eidx   = (const int*)d_in[3];

  float* out = (float*)d_out;
  float* wsf = (float*)d_ws;

  const int N = in_sizes[0] / DIM;      // 100000
  const int E = in_sizes[3] / 2;        // 1600000

  // workspace layout (floats)
  float* wh  = wsf;                     // 64*64
  float* bh  = wsf + 4096;              // 64
  float* y2  = wsf + 4160;              // 1
  float* deg = wsf + 4224;              // N
  float* h   = wsf + 4224 + ((N + 15) & ~15);  // N*64

  hgc_prep_kernel<<<1, 64, 0, stream>>>(weight, bias, wh, bh, y2);

  int tblocks = (N + 15) / 16;
  hgc_transform_kernel<<<tblocks, 128, 0, stream>>>(x, wh, bh, y2, h, N);

  hgc_zero_kernel<<<(N * DIM + 255) / 256, 256, 0, stream>>>(out, N * DIM);
  hgc_zero_kernel<<<(N + 255) / 256, 256, 0, stream>>>(deg, N);

  size_t ethreads = (size_t)E * 32;
  hgc_edge_kernel<<<(unsigned)((ethreads + 255) / 256), 256, 0, stream>>>(
      h, eidx, eidx + E, out, deg, E);

  size_t fthreads = (size_t)N * 32;
  hgc_final_kernel<<<(unsigned)((fthreads + 255) / 256), 256, 0, stream>>>(out, deg, N);
}